// KnnThres_27290222198840
// MI455X (gfx1250) — compile-verified
//
#include <hip/hip_runtime.h>
#include <stdint.h>

#define N_PTS    4096
#define K_NEIGH  20
#define THRES    0.5f
#define CAND_T   0.02734375f   /* 112/4096: E[count]=112, P(count<20) ~ e^-60 */
#define CAP      256
#define WPB      8             /* waves (rows) per block */
#define LANES    32            /* wave32 on gfx1250 */

typedef __attribute__((address_space(3))) void  lds_void_t;

// One b128 async op: each lane moves 16B global -> LDS (wave moves 512B), ASYNCcnt-tracked.
// Inline asm sidesteps the builtin's LangAS-qualified signature (not spellable in HIP).
__device__ __forceinline__ void async_copy_b128(const float* gsrc, float* ldst) {
    // generic -> AS3 addrspacecast yields the 32-bit group-segment byte offset,
    // which is exactly what the instruction's VDST (LDS address VGPR) expects.
    unsigned loff = (unsigned)(size_t)(lds_void_t*)(void*)ldst;
    asm volatile("global_load_async_to_lds_b128 %0, %1, off"
                 :: "v"(loff), "v"((unsigned long long)(size_t)gsrc)
                 : "memory");
}

__device__ __forceinline__ void wait_async0() {
#if __has_builtin(__builtin_amdgcn_s_wait_asynccnt)
    __builtin_amdgcn_s_wait_asynccnt(0);
#else
    asm volatile("s_wait_asynccnt 0" ::: "memory");
#endif
    asm volatile("" ::: "memory");   // keep LDS reads after the wait
}

__global__ __launch_bounds__(WPB * LANES)
void knn_thres_kernel(const float* __restrict__ x, int* __restrict__ out) {
    __shared__ __align__(16) float              rowbuf[WPB][N_PTS];   // 128 KB
    __shared__ unsigned long long               cand[WPB][CAP];       // 16 KB
    __shared__ int                              cnt[WPB];

    const int wave = threadIdx.x >> 5;
    const int lane = threadIdx.x & 31;
    const int row  = blockIdx.x * WPB + wave;          // 0 .. B*N-1
    const int self = row & (N_PTS - 1);                // point index within batch
    const float* __restrict__ grow = x + (size_t)row * N_PTS;
    float* rb = rowbuf[wave];

    // ---- 1) async DMA the whole row into LDS (32 outstanding async b128 ops) ----
    #pragma unroll
    for (int j = 0; j < N_PTS / (LANES * 4); ++j) {
        const int off = j * (LANES * 4) + lane * 4;
        async_copy_b128(grow + off, rb + off);
    }
    if (lane == 0) cnt[wave] = 0;
    wait_async0();

    // ---- 2) single filtered scan: append packed keys (bits<<32 | col) of small values ----
    for (int j = 0; j < N_PTS / (LANES * 4); ++j) {
        const int base = j * (LANES * 4) + lane * 4;
        const float4 v4 = *(const float4*)(rb + base);
        const float vv[4] = {v4.x, v4.y, v4.z, v4.w};
        #pragma unroll
        for (int c = 0; c < 4; ++c) {
            if (vv[c] < CAND_T) {
                int pos = atomicAdd(&cnt[wave], 1);
                if (pos < CAP)
                    cand[wave][pos] =
                        ((unsigned long long)__float_as_uint(vv[c]) << 32) |
                        (unsigned)(base + c);
            }
        }
    }
    asm volatile("" ::: "memory");
    const int C = cnt[wave];   // uniform across the wave

    if (C >= K_NEIGH && C <= CAP) {
        // ---- 3) exact rank of each candidate among candidates; ranks<K are the answer,
        //         already in top_k order (value asc, then index asc via packed key) ----
        for (int p = lane; p < C; p += LANES) {
            const unsigned long long kc = cand[wave][p];
            int rank = 0;
            for (int q = 0; q < C; ++q) rank += (cand[wave][q] < kc) ? 1 : 0;
            if (rank < K_NEIGH) {
                const float v  = __uint_as_float((unsigned)(kc >> 32));
                const int  col = (int)(kc & 0xffffffffu);
                out[(size_t)row * K_NEIGH + rank] = (v > THRES) ? self : col;
            }
        }
    } else {
        // ---- exact fallback (statistically never taken): 20x wave-min extraction ----
        for (int t = 0; t < K_NEIGH; ++t) {
            unsigned long long best = ~0ull;
            for (int j = 0; j < N_PTS / (LANES * 4); ++j) {
                const int base = j * (LANES * 4) + lane * 4;
                const float4 v4 = *(const float4*)(rb + base);
                const float vv[4] = {v4.x, v4.y, v4.z, v4.w};
                #pragma unroll
                for (int c = 0; c < 4; ++c) {
                    unsigned long long kk =
                        ((unsigned long long)__float_as_uint(vv[c]) << 32) |
                        (unsigned)(base + c);
                    best = (kk < best) ? kk : best;
                }
            }
            #pragma unroll
            for (int sh = 16; sh > 0; sh >>= 1) {
                unsigned long long o = __shfl_xor(best, sh, LANES);
                best = (o < best) ? o : best;
            }
            const unsigned bits = (unsigned)(best >> 32);
            const int col = (int)(best & 0xffffffffu);
            if (lane == 0) {
                out[(size_t)row * K_NEIGH + t] =
                    (__uint_as_float(bits) > THRES) ? self : col;
                rb[col] = __uint_as_float(0x7f800000u);  // +inf: remove from pool
            }
            asm volatile("" ::: "memory");
        }
    }
}

extern "C" void kernel_launch(void* const* d_in, const int* in_sizes, int n_in,
                              void* d_out, int out_size, void* d_ws, size_t ws_size,
                              hipStream_t stream) {
    (void)n_in; (void)out_size; (void)d_ws; (void)ws_size;
    const float* x = (const float*)d_in[0];
    int* out = (int*)d_out;
    const int rows   = in_sizes[0] / N_PTS;   // 16*4096 = 65536
    const int blocks = rows / WPB;            // 8192
    knn_thres_kernel<<<blocks, WPB * LANES, 0, stream>>>(x, out);
}